// CannyEdgeDetection_24404004176389
// MI455X (gfx1250) — compile-verified
//
#include <hip/hip_runtime.h>
#include <stdint.h>

#define H 512
#define W 512
#define WPR 16                      // 32-bit words per mask row
#define WPR64 8                     // 64-bit words per mask row
#define WORDS_PER_IMG (H * WPR)     // 8192 u32 = 32KB per image
#define WORDS64_PER_IMG (H * WPR64) // 4096 u64 = 32KB per image
#define PIX_PER_IMG (H * W)

// ---------------- CDNA5 feature guards ----------------
#if __has_builtin(__builtin_amdgcn_global_load_async_to_lds_b32) && \
    __has_builtin(__builtin_amdgcn_s_wait_asynccnt)
#define CANNY_USE_ASYNC 1
#endif

#ifdef CANNY_USE_ASYNC
// Builtin's 1st param is an AS(1) int* ("__device__ int *"), 2nd the AS(3) int*.
// Flat->AS1 is bit-identical; low 32 bits of a flat LDS address are the AS3 offset.
__device__ __forceinline__ __attribute__((address_space(1))) int*
as_global(const void* p) {
  return (__attribute__((address_space(1))) int*)(unsigned long long)(uintptr_t)p;
}
__device__ __forceinline__ __attribute__((address_space(3))) int*
as_lds(void* p) {
  return (__attribute__((address_space(3))) int*)(unsigned)(uintptr_t)p;
}
#endif

__device__ __forceinline__ uint32_t ballot32(bool pred) {
#if __has_builtin(__builtin_amdgcn_ballot_w32)
  return __builtin_amdgcn_ballot_w32(pred);
#else
  return (uint32_t)__ballot(pred);
#endif
}

// ---------------- front end helpers ----------------
__device__ __forceinline__ float gray_of(const float* px, int cstride) {
  float r = px[0], g = px[cstride], b = px[2 * cstride];
  // u8 = floor(clip((x+1)*0.5*256, 0, 255)); pow2 muls are exact == *128
  r = floorf(fminf(fmaxf((r + 1.0f) * 128.0f, 0.0f), 255.0f));
  g = floorf(fminf(fmaxf((g + 1.0f) * 128.0f, 0.0f), 255.0f));
  b = floorf(fminf(fmaxf((b + 1.0f) * 128.0f, 0.0f), 255.0f));
  // reference's left-to-right f32 sum, no FMA contraction
  float y = __fadd_rn(__fadd_rn(__fmul_rn(0.299f, r), __fmul_rn(0.587f, g)),
                      __fmul_rn(0.114f, b));
  return rintf(y);  // RNE == jnp.round
}

// =====================================================================
// Kernel 1: gray -> Sobel -> NMS -> thresholds, bit-packed via wave32 ballot
// block = (32, 8); tile = 32x32 pixels; grid = (16, 16, 32)
// =====================================================================
__global__ __launch_bounds__(256)
void canny_front(const float* __restrict__ x,
                 uint32_t* __restrict__ gstrong,
                 uint32_t* __restrict__ gweak) {
  __shared__ float g[36 * 36];  // gray tile, halo 2, replicate-clamped
  __shared__ float m[34 * 34];  // |grad|_1 tile, halo 1, zero outside image

  const int bx = blockIdx.x;            // word column (w0/32)
  const int by = blockIdx.y;
  const int b  = blockIdx.z;
  const int w0 = bx * 32, h0 = by * 32;
  const int tx = threadIdx.x, ty = threadIdx.y;
  const int tid = ty * 32 + tx;

  const float* img = x + (size_t)b * 3 * PIX_PER_IMG;
#if __has_builtin(__builtin_prefetch)
  __builtin_prefetch(img + (size_t)h0 * W + w0, 0, 0);              // global_prefetch_b8
  __builtin_prefetch(img + PIX_PER_IMG + (size_t)h0 * W + w0, 0, 0);
  __builtin_prefetch(img + 2 * PIX_PER_IMG + (size_t)h0 * W + w0, 0, 0);
#endif

  // gray tile (replicate border, as cv2 Sobel uses)
  for (int idx = tid; idx < 36 * 36; idx += 256) {
    int ly = idx / 36, lx = idx - ly * 36;
    int gh = h0 - 2 + ly; gh = gh < 0 ? 0 : (gh > H - 1 ? H - 1 : gh);
    int gw = w0 - 2 + lx; gw = gw < 0 ? 0 : (gw > W - 1 ? W - 1 : gw);
    g[idx] = gray_of(img + (size_t)gh * W + gw, PIX_PER_IMG);
  }
  __syncthreads();

  // L1 gradient magnitude tile (zero padding outside image, for NMS lookups)
  for (int idx = tid; idx < 34 * 34; idx += 256) {
    int my = idx / 34, mx = idx - my * 34;
    int hh = h0 - 1 + my, ww = w0 - 1 + mx;
    float mv = 0.0f;
    if ((unsigned)hh < (unsigned)H && (unsigned)ww < (unsigned)W) {
      const float* c = &g[(my + 1) * 36 + (mx + 1)];
      float a  = c[-36 - 1], bq = c[-36], cq = c[-36 + 1];
      float d  = c[-1],                  e  = c[+1];
      float f  = c[36 - 1],  gq = c[36], hq = c[36 + 1];
      float gx = (cq + 2.0f * e + hq) - (a + 2.0f * d + f);
      float gy = (f + 2.0f * gq + hq) - (a + 2.0f * bq + cq);
      mv = fabsf(gx) + fabsf(gy);
    }
    m[idx] = mv;
  }
  __syncthreads();

  // per-pixel NMS + double threshold; pack 32 pixels/row with wave32 ballot
  const size_t wbase = (size_t)b * WORDS_PER_IMG + bx;
  #pragma unroll
  for (int i = 0; i < 4; ++i) {
    const int py = ty + 8 * i;  // all 32 lanes of the wave share py
    const float* c = &g[(py + 2) * 36 + (tx + 2)];
    float a  = c[-36 - 1], bq = c[-36], cq = c[-36 + 1];
    float d  = c[-1],                  e  = c[+1];
    float f  = c[36 - 1],  gq = c[36], hq = c[36 + 1];
    float gx = (cq + 2.0f * e + hq) - (a + 2.0f * d + f);
    float gy = (f + 2.0f * gq + hq) - (a + 2.0f * bq + cq);

    const float* mc = &m[(py + 1) * 34 + (tx + 1)];
    float mag = mc[0];
    float ang = atan2f(gy, gx) * 57.29577951308232f;  // degrees(atan2)
    if (ang < 0.0f) ang += 180.0f;
    float q1, q2;
    if (ang < 22.5f || ang >= 157.5f) { q1 = mc[1];        q2 = mc[-1];       }
    else if (ang < 67.5f)             { q1 = mc[34 + 1];   q2 = mc[-34 - 1];  }
    else if (ang < 112.5f)            { q1 = mc[34];       q2 = mc[-34];      }
    else                              { q1 = mc[-34 + 1];  q2 = mc[34 - 1];   }
    float nms = (mag >= q1 && mag >= q2) ? mag : 0.0f;

    uint32_t sm = ballot32(nms > 85.0f);  // HIGH
    uint32_t wm = ballot32(nms > 40.0f);  // LOW
    if (tx == 0) {
      size_t wi = wbase + (size_t)(h0 + py) * WPR;
      gstrong[wi] = sm;
      gweak[wi]   = wm;
    }
  }
}

// =====================================================================
// Kernel 2: 100-iteration hysteresis entirely in LDS, bit-parallel in u64.
// 1 block per image, 1024 threads, 96KB dynamic LDS (weak + ping/pong).
// Lane map: {strip: t>>5, row-in-quad: (t>>3)&3, word: t&7} -> each
// ds_load_b64/ds_store_b64 of a wave touches 4 consecutive rows x 8 u64
// = 64 distinct 4B banks: conflict-free on the 64-bank LDS.
// =====================================================================
__device__ __forceinline__ uint64_t hz64(const uint64_t* S, int r, int c) {
  if ((unsigned)r >= (unsigned)H) return 0ull;  // zero padding outside image
  const uint64_t* row = S + r * WPR64;
  uint64_t v  = row[c];
  uint64_t l  = (c > 0)          ? row[c - 1] : 0ull;
  uint64_t rr = (c < WPR64 - 1)  ? row[c + 1] : 0ull;
  return v | (v << 1) | (l >> 63) | (v >> 1) | (rr << 63);
}

__global__ __launch_bounds__(1024)
void canny_hyst(const uint32_t* __restrict__ gstrong,
                const uint32_t* __restrict__ gweak,
                float* __restrict__ out) {
  extern __shared__ uint64_t smem[];
  uint64_t* Wk = smem;                        // weak mask   (4096 u64)
  uint64_t* A  = smem + WORDS64_PER_IMG;      // strong ping (4096 u64)
  uint64_t* Bf = smem + 2 * WORDS64_PER_IMG;  // strong pong (4096 u64)

  const int b = blockIdx.x;
  const int t = threadIdx.x;
  const size_t base = (size_t)b * WORDS_PER_IMG;

  // Load both masks into LDS via async global->LDS (fallback: plain loads)
  uint32_t* Wk32 = (uint32_t*)Wk;
  uint32_t* A32  = (uint32_t*)A;
  for (int i = t; i < WORDS_PER_IMG; i += 1024) {
#ifdef CANNY_USE_ASYNC
    __builtin_amdgcn_global_load_async_to_lds_b32(as_global(gweak + base + i),
                                                  as_lds(Wk32 + i), 0, 0);
    __builtin_amdgcn_global_load_async_to_lds_b32(as_global(gstrong + base + i),
                                                  as_lds(A32 + i), 0, 0);
#else
    Wk32[i] = gweak[base + i];
    A32[i]  = gstrong[base + i];
#endif
  }
#ifdef CANNY_USE_ASYNC
  __builtin_amdgcn_s_wait_asynccnt(0);
#endif
  __syncthreads();

  // conflict-free ownership: 4 rows at stride 4 inside a 16-row strip
  const int c     = t & (WPR64 - 1);        // u64 word within row
  const int rq    = (t >> 3) & 3;           // row within quad
  const int strip = t >> 5;                 // 32 strips of 16 rows
  const int rbase = strip * 16 + rq;        // rows rbase + 4k, k = 0..3

  uint64_t wk[4];
  #pragma unroll
  for (int k = 0; k < 4; ++k) wk[k] = Wk[(rbase + 4 * k) * WPR64 + c];

  uint64_t* S = A;
  uint64_t* D = Bf;
  for (int it = 0; it < 100; ++it) {  // HYST_ITERS, Jacobi to match reference
    #pragma unroll
    for (int k = 0; k < 4; ++k) {
      const int r = rbase + 4 * k;
      uint64_t dil = hz64(S, r - 1, c) | hz64(S, r, c) | hz64(S, r + 1, c);
      D[r * WPR64 + c] = wk[k] & dil;   // weak & dilate3x3(s)
    }
    __syncthreads();
    uint64_t* tmp = S; S = D; D = tmp;
  }

  // expand final bits to {0,255} floats with b128 stores, coalesced
  const uint32_t* S32 = (const uint32_t*)S;
  float4* o4 = (float4*)(out + (size_t)b * PIX_PER_IMG);
  for (int q = t; q < PIX_PER_IMG / 4; q += 1024) {
    uint32_t wv = S32[q >> 3];       // 8 pixel-quads per u32 word
    int sh = (q & 7) * 4;
    float4 v;
    v.x = ((wv >> (sh + 0)) & 1u) ? 255.0f : 0.0f;
    v.y = ((wv >> (sh + 1)) & 1u) ? 255.0f : 0.0f;
    v.z = ((wv >> (sh + 2)) & 1u) ? 255.0f : 0.0f;
    v.w = ((wv >> (sh + 3)) & 1u) ? 255.0f : 0.0f;
    o4[q] = v;
  }
}

// =====================================================================
extern "C" void kernel_launch(void* const* d_in, const int* in_sizes, int n_in,
                              void* d_out, int out_size, void* d_ws, size_t ws_size,
                              hipStream_t stream) {
  (void)in_sizes; (void)n_in; (void)out_size; (void)ws_size;
  const float* x = (const float*)d_in[0];
  float* out = (float*)d_out;

  uint32_t* gstrong = (uint32_t*)d_ws;                    // 1 MB
  uint32_t* gweak   = gstrong + 32 * WORDS_PER_IMG;       // 1 MB

  dim3 grid1(W / 32, H / 32, 32), blk1(32, 8);
  canny_front<<<grid1, blk1, 0, stream>>>(x, gstrong, gweak);

  size_t shbytes = 3 * WORDS64_PER_IMG * sizeof(uint64_t);  // 96 KB (CDNA5 LDS)
  canny_hyst<<<32, 1024, shbytes, stream>>>(gstrong, gweak, out);
}